// SinkhornProjection_25950192403060
// MI455X (gfx1250) — compile-verified
//
#include <hip/hip_runtime.h>

#define SINKHORN_ITERS 20
#define EPS 1e-6f
#define N 128

typedef unsigned int u32;
typedef __attribute__((ext_vector_type(4))) unsigned int v4u;
typedef __attribute__((ext_vector_type(8))) int v8i;
typedef __attribute__((ext_vector_type(4))) int v4i;

#ifdef __has_builtin
#if __has_builtin(__builtin_amdgcn_tensor_load_to_lds)
#define HAVE_TDM 1
#endif
#endif

// ds_swizzle group-of-32 patterns (no LDS memory traffic, pure lane exchange)
// imm = xor<<10 | or<<5 | and ; and=0x1f keeps it inside the 32-lane group.
template <int IMM>
__device__ __forceinline__ float swz(float x) {
    return __int_as_float(__builtin_amdgcn_ds_swizzle(__float_as_int(x), IMM));
}
#define SWAPX1  0x041f
#define SWAPX2  0x081f
#define SWAPX4  0x101f
#define SWAPX8  0x201f
#define SWAPX16 0x401f

__global__ __launch_bounds__(256)
void sinkhorn_kernel(const float* __restrict__ x, float* __restrict__ out) {
    __shared__ float sM[N * N];    // staging target for TDM (64 KiB)
    __shared__ float sV[N];        // column scaling vector v
    __shared__ float sP[8 * N];    // per-wave column partial sums

    const int t    = threadIdx.x;
    const int tc   = t & 15;        // column-tile index 0..15
    const int tr   = t >> 4;        // row-tile index 0..15
    const int lane = t & 31;
    const int w    = t >> 5;        // wave index 0..7
    const int r0   = tr * 8;
    const int c0   = tc * 8;

    const float* gx   = x   + (long long)blockIdx.x * (N * N);
    float*       gout = out + (long long)blockIdx.x * (N * N);

    // ---- Stage the whole 128x128 matrix into LDS via the Tensor Data Mover ----
#ifdef HAVE_TDM
    if (w == 0) {
        unsigned long long ga = (unsigned long long)gx;
        u32 lds_off = (u32)(unsigned long long)&sM[0];
        v4u g0;
        g0[0] = 1u;                                  // count=1, no gather
        g0[1] = lds_off;                             // lds_addr
        g0[2] = (u32)(ga & 0xffffffffull);           // global_addr[31:0]
        g0[3] = (u32)((ga >> 32) & 0x1ffffffull)     // global_addr[56:32]
              | 0x80000000u;                         // type=2 (image)
        v8i g1;
        g1[0] = (int)(2u << 16);      // data_size=4B, wg_mask=0
        g1[1] = (int)(128u << 16);    // tensor_dim0 = 128
        g1[2] = (int)(128u << 16);    // tensor_dim1 = 128
        g1[3] = (int)(128u << 16);    // tile_dim0 = 128
        g1[4] = (int)(128u);          // tile_dim1 = 128, tile_dim2 = 0
        g1[5] = (int)(128u);          // tensor_dim0_stride = 128
        g1[6] = 0;
        g1[7] = 0;
        v4i g2 = (v4i)0;
        v4i g3 = (v4i)0;
        v8i g4 = (v8i)0;
        __builtin_amdgcn_tensor_load_to_lds(g0, g1, g2, g3, g4, 0);
        __builtin_amdgcn_s_wait_tensorcnt(0);
    }
#else
    for (int idx = t; idx < N * N / 4; idx += 256)
        ((float4*)sM)[idx] = ((const float4*)gx)[idx];
#endif
    if (t < N) sV[t] = 1.0f;
    __syncthreads();

    // ---- Pull private 8x8 tile into registers ----
    float m[8][8];
#pragma unroll
    for (int i = 0; i < 8; ++i) {
        float4 a = *(const float4*)&sM[(r0 + i) * N + c0];
        float4 b = *(const float4*)&sM[(r0 + i) * N + c0 + 4];
        m[i][0] = a.x; m[i][1] = a.y; m[i][2] = a.z; m[i][3] = a.w;
        m[i][4] = b.x; m[i][5] = b.y; m[i][6] = b.z; m[i][7] = b.w;
    }

    // ---- rowmax (butterfly over the 16-lane row group) then exp ----
    float mx[8];
#pragma unroll
    for (int i = 0; i < 8; ++i) {
        float v = m[i][0];
#pragma unroll
        for (int j = 1; j < 8; ++j) v = fmaxf(v, m[i][j]);
        v = fmaxf(v, swz<SWAPX1>(v));
        v = fmaxf(v, swz<SWAPX2>(v));
        v = fmaxf(v, swz<SWAPX4>(v));
        v = fmaxf(v, swz<SWAPX8>(v));
        mx[i] = v;
    }
#pragma unroll
    for (int i = 0; i < 8; ++i)
#pragma unroll
        for (int j = 0; j < 8; ++j)
            m[i][j] = __expf(m[i][j] - mx[i]);

    float u[8];
#pragma unroll
    for (int i = 0; i < 8; ++i) u[i] = 1.0f;

    // ---- Sinkhorn iterations on scaling vectors only ----
    for (int it = 0; it < SINKHORN_ITERS; ++it) {
        // row pass: rs_i = sum_c M[r][c] * v[c]
        float4 va = *(const float4*)&sV[c0];
        float4 vb = *(const float4*)&sV[c0 + 4];
        float vj[8] = {va.x, va.y, va.z, va.w, vb.x, vb.y, vb.z, vb.w};
        float rs[8];
#pragma unroll
        for (int i = 0; i < 8; ++i) {
            float s = 0.0f;
#pragma unroll
            for (int j = 0; j < 8; ++j) s = fmaf(m[i][j], vj[j], s);
            s += swz<SWAPX1>(s);
            s += swz<SWAPX2>(s);
            s += swz<SWAPX4>(s);
            s += swz<SWAPX8>(s);
            rs[i] = s;
        }
#pragma unroll
        for (int i = 0; i < 8; ++i)
            u[i] = u[i] / (u[i] * rs[i] + EPS);   // row normalize -> u update

        // col pass: cs_j = sum_r u[r] * M[r][c]
        float cs[8];
#pragma unroll
        for (int j = 0; j < 8; ++j) {
            float s = 0.0f;
#pragma unroll
            for (int i = 0; i < 8; ++i) s = fmaf(u[i], m[i][j], s);
            s += swz<SWAPX16>(s);                 // combine the 2 row-groups in this wave
            cs[j] = s;
        }
        if (lane < 16) {                          // one copy per wave
            float4 p0 = {cs[0], cs[1], cs[2], cs[3]};
            float4 p1 = {cs[4], cs[5], cs[6], cs[7]};
            *(float4*)&sP[w * N + c0]     = p0;
            *(float4*)&sP[w * N + c0 + 4] = p1;
        }
        __syncthreads();
        if (t < N) {                              // reduce 8 wave partials, update v
            float s = 0.0f;
#pragma unroll
            for (int ww = 0; ww < 8; ++ww) s += sP[ww * N + t];
            float vv = sV[t];
            sV[t] = vv / (vv * s + EPS);          // col normalize -> v update
        }
        __syncthreads();
    }

    // ---- Materialize result: out = diag(u) * M * diag(v) ----
    {
        float4 va = *(const float4*)&sV[c0];
        float4 vb = *(const float4*)&sV[c0 + 4];
        float vj[8] = {va.x, va.y, va.z, va.w, vb.x, vb.y, vb.z, vb.w};
#pragma unroll
        for (int i = 0; i < 8; ++i) {
            float ui = u[i];
            float4 o0 = {ui * m[i][0] * vj[0], ui * m[i][1] * vj[1],
                         ui * m[i][2] * vj[2], ui * m[i][3] * vj[3]};
            float4 o1 = {ui * m[i][4] * vj[4], ui * m[i][5] * vj[5],
                         ui * m[i][6] * vj[6], ui * m[i][7] * vj[7]};
            *(float4*)&gout[(r0 + i) * N + c0]     = o0;
            *(float4*)&gout[(r0 + i) * N + c0 + 4] = o1;
        }
    }
}

extern "C" void kernel_launch(void* const* d_in, const int* in_sizes, int n_in,
                              void* d_out, int out_size, void* d_ws, size_t ws_size,
                              hipStream_t stream) {
    (void)n_in; (void)out_size; (void)d_ws; (void)ws_size;
    const float* x = (const float*)d_in[0];
    float* out = (float*)d_out;
    int batches = in_sizes[0] / (N * N);   // 4096
    sinkhorn_kernel<<<dim3(batches), dim3(256), 0, stream>>>(x, out);
}